// GCN_9062380994846
// MI455X (gfx1250) — compile-verified
//
#include <hip/hip_runtime.h>
#include <hip/hip_bf16.h>

typedef float v2f __attribute__((ext_vector_type(2)));
typedef float v8f __attribute__((ext_vector_type(8)));

#define N_NODES   20000
#define N_EDGES   320000
#define DIM       256
#define N_GRAPHS  64
#define N_CLASSES 10

// ---------------- degree / norm ----------------
__global__ void fill_deg(float* deg, int n) {
    int i = blockIdx.x * blockDim.x + threadIdx.x;
    if (i < n) deg[i] = 1.0f;                       // self-loop
}
__global__ void edge_deg(const int* __restrict__ dst, float* deg, int e) {
    int i = blockIdx.x * blockDim.x + threadIdx.x;
    if (i < e) atomicAdd(&deg[dst[i]], 1.0f);
}
__global__ void deg_to_dinv(float* deg, int n) {
    int i = blockIdx.x * blockDim.x + threadIdx.x;
    if (i < n) deg[i] = 1.0f / sqrtf(deg[i]);       // deg >= 1 always
}

// ---------------- fp32 WMMA GEMM: H[M,256] = X[M,256] @ W[256,256] ----------------
// One wave computes a 32-row x 64-col strip (8 WMMA tiles): two A fragments share
// each set of 4 B fragments -> 8 WMMAs per 10 loads per k-step.
__global__ __launch_bounds__(128) void gemm_xw(const float* __restrict__ X,
                                               const float* __restrict__ W,
                                               float* __restrict__ H) {
    const int wave = threadIdx.x >> 5;
    const int lane = threadIdx.x & 31;
    const int tile = blockIdx.x * 4 + wave;   // 0 .. (M/32)*4 - 1
    const int m0   = (tile >> 2) * 32;
    const int n0   = (tile & 3) * 64;
    const int lrow = lane & 15;               // m (A), n (B/D) sub-index
    const int lhi  = lane >> 4;               // selects K pair / M half

    v8f acc0 = {}, acc1 = {}, acc2 = {}, acc3 = {};   // rows m0 .. m0+15
    v8f acc4 = {}, acc5 = {}, acc6 = {}, acc7 = {};   // rows m0+16 .. m0+31
    const float* xrow0 = X + (size_t)(m0 + lrow) * DIM;
    const float* xrow1 = X + (size_t)(m0 + 16 + lrow) * DIM;

#pragma unroll 4
    for (int k = 0; k < DIM; k += 4) {
        const int kk = k + 2 * lhi;
        // A fragments: X[m][kk], X[m][kk+1] (8B-aligned pair per lane)
        v2f a0, a1;
        {
            const float2 ax0 = *(const float2*)(xrow0 + kk);
            const float2 ax1 = *(const float2*)(xrow1 + kk);
            a0.x = ax0.x; a0.y = ax0.y;
            a1.x = ax1.x; a1.y = ax1.y;
        }
        // B fragments: W[kk][n0+16j+lrow], W[kk+1][n0+16j+lrow]
        const float* wb = W + (size_t)kk * DIM + n0 + lrow;
        v2f b0, b1, b2, b3;
        b0.x = wb[0];  b0.y = wb[DIM];
        b1.x = wb[16]; b1.y = wb[16 + DIM];
        b2.x = wb[32]; b2.y = wb[32 + DIM];
        b3.x = wb[48]; b3.y = wb[48 + DIM];

        acc0 = __builtin_amdgcn_wmma_f32_16x16x4_f32(false, a0, false, b0, (short)0, acc0, false, false);
        acc1 = __builtin_amdgcn_wmma_f32_16x16x4_f32(false, a0, false, b1, (short)0, acc1, false, false);
        acc2 = __builtin_amdgcn_wmma_f32_16x16x4_f32(false, a0, false, b2, (short)0, acc2, false, false);
        acc3 = __builtin_amdgcn_wmma_f32_16x16x4_f32(false, a0, false, b3, (short)0, acc3, false, false);
        acc4 = __builtin_amdgcn_wmma_f32_16x16x4_f32(false, a1, false, b0, (short)0, acc4, false, false);
        acc5 = __builtin_amdgcn_wmma_f32_16x16x4_f32(false, a1, false, b1, (short)0, acc5, false, false);
        acc6 = __builtin_amdgcn_wmma_f32_16x16x4_f32(false, a1, false, b2, (short)0, acc6, false, false);
        acc7 = __builtin_amdgcn_wmma_f32_16x16x4_f32(false, a1, false, b3, (short)0, acc7, false, false);
    }

    // D layout: vgpr v -> row base + v + 8*lhi, lane%16 -> col
#pragma unroll
    for (int v = 0; v < 8; ++v) {
        const size_t mA = (size_t)(m0 + v + 8 * lhi);
        const size_t mB = mA + 16;
        float* hrA = H + mA * DIM + n0 + lrow;
        float* hrB = H + mB * DIM + n0 + lrow;
        hrA[0]  = acc0[v];  hrA[16] = acc1[v];  hrA[32] = acc2[v];  hrA[48] = acc3[v];
        hrB[0]  = acc4[v];  hrB[16] = acc5[v];  hrB[32] = acc6[v];  hrB[48] = acc7[v];
    }
}

// ---------------- aggregation ----------------
// out[i][c] = bias[c] + h[i][c]*dinv[i]^2 (+ res[i][c])   -- self-loop + bias + residual
__global__ void agg_init(const float* __restrict__ h, const float* __restrict__ res,
                         const float* __restrict__ bias, const float* __restrict__ dinv,
                         float* __restrict__ out) {
    int tid = blockIdx.x * blockDim.x + threadIdx.x;   // N_NODES*256 threads
    int i = tid >> 8, c = tid & 255;
    if (i >= N_NODES) return;
    float di = dinv[i];
    float v = bias[c] + h[tid] * di * di;
    if (res) v += res[tid];
    out[tid] = v;
}

// one edge handled by 64 lanes, 4 channels (float4) each
__global__ __launch_bounds__(256) void edge_scatter(const float* __restrict__ h,
                                                    const int* __restrict__ src,
                                                    const int* __restrict__ dst,
                                                    const float* __restrict__ dinv,
                                                    float* __restrict__ out) {
    long tid = (long)blockIdx.x * blockDim.x + threadIdx.x;
    int e  = (int)(tid >> 6);
    int c4 = (int)(tid & 63) << 2;
    if (e >= N_EDGES) return;
    int s = src[e], d = dst[e];
    float nrm = dinv[s] * dinv[d];
    float4 hv = *(const float4*)(h + (size_t)s * DIM + c4);
    float* o = out + (size_t)d * DIM + c4;
    atomicAdd(o + 0, hv.x * nrm);
    atomicAdd(o + 1, hv.y * nrm);
    atomicAdd(o + 2, hv.z * nrm);
    atomicAdd(o + 3, hv.w * nrm);
}

__global__ void relu_inplace(float* __restrict__ x, int n) {
    int i = blockIdx.x * blockDim.x + threadIdx.x;
    if (i < n) x[i] = fmaxf(x[i], 0.0f);
}

// ---------------- pooling + classifier ----------------
__global__ void zero_buf(float* p, int n) {
    int i = blockIdx.x * blockDim.x + threadIdx.x;
    if (i < n) p[i] = 0.0f;
}
__global__ void count_nodes(const int* __restrict__ batch, float* counts, int n) {
    int i = blockIdx.x * blockDim.x + threadIdx.x;
    if (i < n) atomicAdd(&counts[batch[i]], 1.0f);
}
// batch is sorted: each block sums 16 consecutive nodes per channel, flushing on segment change
__global__ __launch_bounds__(256) void pool_sum(const float* __restrict__ h3,
                                                const int* __restrict__ batch,
                                                float* __restrict__ pool) {
    int c  = threadIdx.x;           // channel
    int i0 = blockIdx.x * 16;       // node range [i0, i0+16)
    int g  = batch[i0];
    float s = 0.0f;
    for (int i = i0; i < i0 + 16 && i < N_NODES; ++i) {
        int gi = batch[i];
        if (gi != g) { atomicAdd(&pool[(size_t)g * DIM + c], s); s = 0.0f; g = gi; }
        s += h3[(size_t)i * DIM + c];
    }
    atomicAdd(&pool[(size_t)g * DIM + c], s);
}

__global__ __launch_bounds__(64) void classifier(const float* __restrict__ pool,
                                                 const float* __restrict__ counts,
                                                 const float* __restrict__ Wc,
                                                 const float* __restrict__ bc,
                                                 float* __restrict__ out) {
    int g = threadIdx.x;
    if (g >= N_GRAPHS) return;
    float inv = 1.0f / fmaxf(counts[g], 1.0f);
    float lg[N_CLASSES];
#pragma unroll
    for (int k = 0; k < N_CLASSES; ++k) lg[k] = bc[k];
    for (int c = 0; c < DIM; ++c) {
        float p = pool[(size_t)g * DIM + c] * inv;
#pragma unroll
        for (int k = 0; k < N_CLASSES; ++k) lg[k] += p * Wc[c * N_CLASSES + k];
    }
    float mx = lg[0];
#pragma unroll
    for (int k = 1; k < N_CLASSES; ++k) mx = fmaxf(mx, lg[k]);
    float se = 0.0f;
#pragma unroll
    for (int k = 0; k < N_CLASSES; ++k) se += expf(lg[k] - mx);
    float lse = mx + logf(se);
#pragma unroll
    for (int k = 0; k < N_CLASSES; ++k) {
        out[g * N_CLASSES + k] = lg[k];                                   // logits
        out[N_GRAPHS * N_CLASSES + g * N_CLASSES + k] = lg[k] - lse;      // log_softmax
    }
}

// ---------------- launch ----------------
extern "C" void kernel_launch(void* const* d_in, const int* in_sizes, int n_in,
                              void* d_out, int out_size, void* d_ws, size_t ws_size,
                              hipStream_t stream) {
    const float* x     = (const float*)d_in[0];
    const int*   eidx  = (const int*)d_in[1];
    const int*   batch = (const int*)d_in[2];
    const float* W1 = (const float*)d_in[3]; const float* b1 = (const float*)d_in[4];
    const float* W2 = (const float*)d_in[5]; const float* b2 = (const float*)d_in[6];
    const float* W3 = (const float*)d_in[7]; const float* b3 = (const float*)d_in[8];
    const float* Wc = (const float*)d_in[9]; const float* bc = (const float*)d_in[10];
    const int* src = eidx;
    const int* dst = eidx + N_EDGES;
    float* out = (float*)d_out;

    // workspace layout (256B aligned regions)
    char* ws = (char*)d_ws;
    size_t off = 0;
    auto carve = [&](size_t bytes) { float* p = (float*)(ws + off); off = (off + bytes + 255) & ~(size_t)255; return p; };
    float* dinv   = carve((size_t)N_NODES * 4);
    float* bufH   = carve((size_t)N_NODES * DIM * 4);   // GEMM output per layer
    float* buf1   = carve((size_t)N_NODES * DIM * 4);   // h1, reused for h3
    float* buf2   = carve((size_t)N_NODES * DIM * 4);   // h2
    float* pool   = carve((size_t)N_GRAPHS * DIM * 4);
    float* counts = carve((size_t)N_GRAPHS * 4);
    (void)ws_size; (void)n_in; (void)in_sizes; (void)out_size;

    const int NE_ELEMS = N_NODES * DIM;                 // 5,120,000
    const int GEMM_BLOCKS = (N_NODES / 32);             // 625 blocks x 4 waves (4 n-groups each)

    // degree -> dinv (graph-invariant, computed once)
    fill_deg<<<(N_NODES + 255) / 256, 256, 0, stream>>>(dinv, N_NODES);
    edge_deg<<<(N_EDGES + 255) / 256, 256, 0, stream>>>(dst, dinv, N_EDGES);
    deg_to_dinv<<<(N_NODES + 255) / 256, 256, 0, stream>>>(dinv, N_NODES);

    // layer 1: h1 = relu(gcn(x, W1, b1))
    gemm_xw<<<GEMM_BLOCKS, 128, 0, stream>>>(x, W1, bufH);
    agg_init<<<(NE_ELEMS + 255) / 256, 256, 0, stream>>>(bufH, nullptr, b1, dinv, buf1);
    edge_scatter<<<(N_EDGES * 64) / 256, 256, 0, stream>>>(bufH, src, dst, dinv, buf1);
    relu_inplace<<<(NE_ELEMS + 255) / 256, 256, 0, stream>>>(buf1, NE_ELEMS);

    // layer 2: h2 = relu(h1 + gcn(h1, W2, b2))
    gemm_xw<<<GEMM_BLOCKS, 128, 0, stream>>>(buf1, W2, bufH);
    agg_init<<<(NE_ELEMS + 255) / 256, 256, 0, stream>>>(bufH, buf1, b2, dinv, buf2);
    edge_scatter<<<(N_EDGES * 64) / 256, 256, 0, stream>>>(bufH, src, dst, dinv, buf2);
    relu_inplace<<<(NE_ELEMS + 255) / 256, 256, 0, stream>>>(buf2, NE_ELEMS);

    // layer 3: h3 = gcn(h2, W3, b3)   (h3 -> buf1, reuse)
    gemm_xw<<<GEMM_BLOCKS, 128, 0, stream>>>(buf2, W3, bufH);
    agg_init<<<(NE_ELEMS + 255) / 256, 256, 0, stream>>>(bufH, nullptr, b3, dinv, buf1);
    edge_scatter<<<(N_EDGES * 64) / 256, 256, 0, stream>>>(bufH, src, dst, dinv, buf1);

    // mean pool + classifier + log_softmax
    zero_buf<<<(N_GRAPHS * DIM + 255) / 256, 256, 0, stream>>>(pool, N_GRAPHS * DIM);
    zero_buf<<<1, 64, 0, stream>>>(counts, N_GRAPHS);
    count_nodes<<<(N_NODES + 255) / 256, 256, 0, stream>>>(batch, counts, N_NODES);
    pool_sum<<<(N_NODES + 15) / 16, 256, 0, stream>>>(buf1, batch, pool);
    classifier<<<1, 64, 0, stream>>>(pool, counts, Wc, bc, out);
}